// GraphLearner_10058813407527
// MI455X (gfx1250) — compile-verified
//
#include <hip/hip_runtime.h>
#include <math.h>

// ---------------- constants ----------------
#define NODES   1024
#define NN      (NODES * NODES)          // 1,048,576
#define FDIM    64
#define EMBD    32
#define PER     12
#define ORD     3
#define BB      64
#define TT      12
#define DD      8
#define BD      (BB * DD)                // 512
#define NBD     (NODES * BD)             // 524,288
#define TM1     (TT - 1)                 // 11

typedef __attribute__((ext_vector_type(2))) float v2f;
typedef __attribute__((ext_vector_type(8))) float v8f;

// ---------------- coeff = softmax(emb @ W_c) ----------------
__global__ void coeff_kernel(const float* __restrict__ emb,
                             const float* __restrict__ W_c,
                             float* __restrict__ coeff) {
    int p = threadIdx.x;
    if (p >= PER) return;
    float a0 = 0.f, a1 = 0.f, a2 = 0.f;
    for (int e = 0; e < EMBD; ++e) {
        float v = emb[p * EMBD + e];
        a0 += v * W_c[e * ORD + 0];
        a1 += v * W_c[e * ORD + 1];
        a2 += v * W_c[e * ORD + 2];
    }
    float mx = fmaxf(a0, fmaxf(a1, a2));
    float e0 = expf(a0 - mx), e1 = expf(a1 - mx), e2 = expf(a2 - mx);
    float s = e0 + e1 + e2;
    coeff[p * ORD + 0] = e0 / s;
    coeff[p * ORD + 1] = e1 / s;
    coeff[p * ORD + 2] = e2 / s;
}

// ---------------- sc[t][n][b*8+d] = inputs[b][n][t][d] ----------------
__global__ void sc_kernel(const float* __restrict__ in, float* __restrict__ sc) {
    long idx = (long)blockIdx.x * blockDim.x + threadIdx.x;
    if (idx >= (long)TT * NBD) return;
    int k = (int)(idx & (BD - 1));        // b*8 + d
    int n = (int)((idx >> 9) & (NODES - 1));
    int t = (int)(idx >> 19);
    int b = k >> 3, d = k & 7;
    sc[idx] = in[((((long)b * NODES + n) * TT + t) << 3) + d];
}

// ---------------- register-blocked fp32 WMMA GEMM ----------------
// Each wave computes a 64x64 block of C as a 4x4 grid of 16x16 WMMA tiles:
// 8 fragment loads feed 16 v_wmma_f32_16x16x4_f32 per K-step (16 FLOP/byte
// from the 192 MB L2, where every operand here is resident).
// Addressing uses 32-bit element offsets off uniform SGPR base pointers to
// keep per-lane address state at 8 scalars; __launch_bounds__(128,1) gives
// the allocator the full VGPR file so the 128 accumulator VGPRs never spill.
// C = scale * op(A[M,K] * B[K,N]), row-major; TRANSB: B[k][j] = Bsrc[j*ldb + k]
// M, N multiples of 64; K multiple of 4; lda/ldb even (v2f-aligned fragments).
template <bool TRANSB, bool RELU>
__global__ __launch_bounds__(128, 1)
void wmma_gemm_f32(const float* __restrict__ A,
                   const float* __restrict__ Bm,
                   float* __restrict__ C,
                   int M, int N, int K,
                   int lda, int ldb, int ldc, float scale,
                   int aStride, int bStride, int cStride) {
    const int lane    = threadIdx.x & 31;
    const int wave    = threadIdx.x >> 5;
    const int blocksN = N >> 6;                                 // 64-wide blocks
    const int blkId   = blockIdx.x * (blockDim.x >> 5) + wave;  // wave-uniform
    if (blkId >= (M >> 6) * blocksN) return;                    // whole wave exits: EXEC stays all-1s

    // uniform (SGPR) batch bases
    const float* Ab = A  + (long)blockIdx.z * aStride;
    const float* Bb = Bm + (long)blockIdx.z * bStride;
    float*       Cb = C  + (long)blockIdx.z * cStride;

    const int tm   = (blkId / blocksN) << 6;
    const int tn   = (blkId % blocksN) << 6;
    const int half = lane >> 4;                 // 0: K pair {0,1}, 1: K pair {2,3}
    const int l15  = lane & 15;

    // 32-bit per-lane element offsets (all < 2^31)
    int aOff[4], bOff[4];
#pragma unroll
    for (int i = 0; i < 4; ++i) aOff[i] = (tm + 16 * i + l15) * lda;
#pragma unroll
    for (int j = 0; j < 4; ++j)
        bOff[j] = TRANSB ? (tn + 16 * j + l15) * ldb : (tn + 16 * j + l15);

    v8f acc[4][4];
#pragma unroll
    for (int i = 0; i < 4; ++i)
#pragma unroll
        for (int j = 0; j < 4; ++j) acc[i][j] = (v8f){};

    for (int k = 0; k < K; k += 4) {
        const int ka = k + 2 * half;
        v2f a[4], b[4];
#pragma unroll
        for (int i = 0; i < 4; ++i)
            a[i] = *(const v2f*)&Ab[aOff[i] + ka];        // one b64 load
#pragma unroll
        for (int j = 0; j < 4; ++j) {
            if (TRANSB) {
                b[j] = *(const v2f*)&Bb[bOff[j] + ka];    // one b64 load
            } else {
                b[j].x = Bb[bOff[j] + ka * ldb];          // K-pairs strided by ldb
                b[j].y = Bb[bOff[j] + (ka + 1) * ldb];
            }
        }
#pragma unroll
        for (int i = 0; i < 4; ++i)
#pragma unroll
            for (int j = 0; j < 4; ++j)
                acc[i][j] = __builtin_amdgcn_wmma_f32_16x16x4_f32(
                    false, a[i], false, b[j], (short)0, acc[i][j], false, false);
    }

#pragma unroll
    for (int i = 0; i < 4; ++i)
#pragma unroll
        for (int j = 0; j < 4; ++j) {
            const int cCol = tn + 16 * j + l15;
#pragma unroll
            for (int v = 0; v < 8; ++v) {
                float val = acc[i][j][v] * scale;
                if (RELU) val = fmaxf(val, 0.f);
                Cb[(tm + 16 * i + v + 8 * half) * ldc + cCol] = val;
            }
        }
}

// ---------------- fused masked-MAE metric per (t, p) ----------------
__global__ void metrics_kernel(const float* __restrict__ sc,
                               const float* __restrict__ Y1,
                               const float* __restrict__ Y2,
                               const float* __restrict__ coeff,
                               float* __restrict__ metric) {
    const int p = blockIdx.x;      // 12
    const int t = blockIdx.y;      // 11
    const float c0 = coeff[p * 3 + 0] * (1.f / 3.f);
    const float c1 = coeff[p * 3 + 1] * (1.f / 3.f);
    const float c2 = coeff[p * 3 + 2] * (1.f / 3.f);
    const float* s0 = sc + (long)t * NBD;
    const float* s1 = sc + (long)(t + 1) * NBD;
    const float* y1 = Y1 + (long)t * NBD;
    const float* y2 = Y2 + (long)t * NBD;

    float num = 0.f, den = 0.f;
    for (long e = threadIdx.x; e < (long)NBD; e += blockDim.x) {
        float tru  = s1[e];
        float m    = (tru != 0.f) ? 1.f : 0.f;
        float pred = c0 * s0[e] + c1 * y1[e] + c2 * y2[e];
        num += fabsf(pred - tru) * m;
        den += m;
    }
    __shared__ float sn[256], sd[256];
    sn[threadIdx.x] = num;
    sd[threadIdx.x] = den;
    __syncthreads();
    for (int s = 128; s > 0; s >>= 1) {
        if (threadIdx.x < s) {
            sn[threadIdx.x] += sn[threadIdx.x + s];
            sd[threadIdx.x] += sd[threadIdx.x + s];
        }
        __syncthreads();
    }
    if (threadIdx.x == 0)
        metric[t * PER + p] = (sd[0] > 0.f) ? sn[0] / sd[0] : 0.f;
}

// ---------------- argmin over p (first-min, like jnp.argmin) ----------------
__global__ void argmin_kernel(const float* __restrict__ metric, int* __restrict__ best) {
    int t = threadIdx.x;
    if (t >= TM1) return;
    int   bi = 0;
    float bv = metric[t * PER];
    for (int p = 1; p < PER; ++p) {
        float v = metric[t * PER + p];
        if (v < bv) { bv = v; bi = p; }
    }
    best[t] = bi;
}

// ---- context slice = graphs[idxPtr[z]] = (c0*I + c1*g + c2*g2)/3 ----
__global__ void build_context_kernel(const float* __restrict__ g,
                                     const float* __restrict__ g2,
                                     const float* __restrict__ coeff,
                                     const int* __restrict__ idxPtr,
                                     float* __restrict__ outBase) {
    long e = (long)blockIdx.x * blockDim.x + threadIdx.x;
    if (e >= (long)NN) return;
    const int t = blockIdx.z;
    const float* c = coeff + idxPtr[t] * 3;
    int i = (int)(e >> 10), j = (int)(e & (NODES - 1));
    float v = c[1] * g[e] + c[2] * g2[e];
    if (i == j) v += c[0];
    outBase[(long)t * NN + e] = v * (1.f / 3.f);
}

// ---------------- xc = sc[T-1] - row_mean ----------------
__global__ void xc_kernel(const float* __restrict__ sc11, float* __restrict__ xc) {
    const int n = blockIdx.x;
    __shared__ float red[256];
    float s = 0.f;
    for (int c = threadIdx.x; c < BD; c += 256) s += sc11[(long)n * BD + c];
    red[threadIdx.x] = s;
    __syncthreads();
    for (int st = 128; st > 0; st >>= 1) {
        if (threadIdx.x < st) red[threadIdx.x] += red[threadIdx.x + st];
        __syncthreads();
    }
    const float mean = red[0] * (1.f / (float)BD);
    for (int c = threadIdx.x; c < BD; c += 256)
        xc[(long)n * BD + c] = sc11[(long)n * BD + c] - mean;
}

// ---------------- Gi[t][r] = W_ih[r] . context[t] ----------------
__global__ void gi_kernel(const float* __restrict__ W_ih,
                          const float* __restrict__ ctx,
                          float* __restrict__ Gi) {
    const int r = blockIdx.x;   // 36
    const int t = blockIdx.y;   // 12
    const float* w = W_ih + (long)r * NN;
    const float* x = ctx + (long)t * NN;
    float s = 0.f;
    for (long e = threadIdx.x; e < (long)NN; e += blockDim.x) s += w[e] * x[e];
    __shared__ float red[256];
    red[threadIdx.x] = s;
    __syncthreads();
    for (int st = 128; st > 0; st >>= 1) {
        if (threadIdx.x < st) red[threadIdx.x] += red[threadIdx.x + st];
        __syncthreads();
    }
    if (threadIdx.x == 0) Gi[t * 36 + r] = red[0];
}

// ---------------- tiny scalar GRU + argmax ----------------
__device__ __forceinline__ float sigf(float x) { return 1.f / (1.f + expf(-x)); }

__global__ void gru_kernel(const float* __restrict__ Gi,
                           const float* __restrict__ W_hh,
                           const float* __restrict__ b_ih,
                           const float* __restrict__ b_hh,
                           int* __restrict__ lastIdx) {
    if (threadIdx.x != 0) return;
    float h[PER];
    for (int j = 0; j < PER; ++j) h[j] = 0.f;
    for (int t = 0; t < TT; ++t) {
        float gi[36], gh[36], hn[PER];
        for (int q = 0; q < 36; ++q) gi[q] = Gi[t * 36 + q] + b_ih[q];
        for (int q = 0; q < 36; ++q) {
            float s = b_hh[q];
            for (int j = 0; j < PER; ++j) s += W_hh[q * PER + j] * h[j];
            gh[q] = s;
        }
        for (int j = 0; j < PER; ++j) {
            float r  = sigf(gi[j] + gh[j]);
            float z  = sigf(gi[PER + j] + gh[PER + j]);
            float nn = tanhf(gi[2 * PER + j] + r * gh[2 * PER + j]);
            hn[j] = (1.f - z) * nn + z * h[j];
        }
        for (int j = 0; j < PER; ++j) h[j] = hn[j];
    }
    int bi = 0;
    for (int j = 1; j < PER; ++j)
        if (h[j] > h[bi]) bi = j;   // argmax(softmax(x)) == argmax(x), first max
    lastIdx[0] = bi;
}

// ---------------- host orchestration ----------------
extern "C" void kernel_launch(void* const* d_in, const int* in_sizes, int n_in,
                              void* d_out, int out_size, void* d_ws, size_t ws_size,
                              hipStream_t stream) {
    (void)in_sizes; (void)n_in; (void)out_size; (void)ws_size;
    const float* inputs = (const float*)d_in[0];
    const float* E      = (const float*)d_in[1];
    const float* emb    = (const float*)d_in[2];
    const float* W_c    = (const float*)d_in[3];
    const float* W_ih   = (const float*)d_in[4];
    const float* W_hh   = (const float*)d_in[5];
    const float* b_ih   = (const float*)d_in[6];
    const float* b_hh   = (const float*)d_in[7];
    float* out = (float*)d_out;                 // [12, 1024, 1024]

    // workspace carve-up (floats)
    float* ws = (float*)d_ws;
    size_t off = 0;
    float* g      = ws + off; off += (size_t)NN;          // 4 MB
    float* g2     = ws + off; off += (size_t)NN;          // 4 MB
    float* sc     = ws + off; off += (size_t)TT  * NBD;   // 24 MB
    float* Y1     = ws + off; off += (size_t)TM1 * NBD;   // 22 MB
    float* Y2     = ws + off; off += (size_t)TM1 * NBD;   // 22 MB
    float* xc     = ws + off; off += (size_t)NBD;         // 2 MB
    float* coeff  = ws + off; off += 64;
    float* metric = ws + off; off += 256;
    float* Gi     = ws + off; off += 512;
    int*   best    = (int*)(ws + off); off += 16;
    int*   lastIdx = (int*)(ws + off); off += 16;

    // 1. coeff = softmax(emb @ W_c)
    coeff_kernel<<<1, 32, 0, stream>>>(emb, W_c, coeff);

    // 2. sc transpose
    {
        long total = (long)TT * NBD;
        sc_kernel<<<(int)((total + 255) / 256), 256, 0, stream>>>(inputs, sc);
    }

    // wave-block geometry: 64x64 per wave, 4 waves (128 threads) per block
    // 1024x1024 -> 256 wave blocks -> 64 thread blocks
    // 1024x512  -> 128 wave blocks -> 32 thread blocks

    // 3. g = relu(E @ E^T)   [1024x1024x64]
    wmma_gemm_f32<true, true><<<dim3(64), 128, 0, stream>>>(
        E, E, g, NODES, NODES, FDIM, FDIM, FDIM, NODES, 1.0f, 0, 0, 0);

    // 4. g2 = g @ g          [1024^3]
    wmma_gemm_f32<false, false><<<dim3(64), 128, 0, stream>>>(
        g, g, g2, NODES, NODES, NODES, NODES, NODES, NODES, 1.0f, 0, 0, 0);

    // 5. Y1[t] = g @ sc[t], Y2[t] = g @ Y1[t]  (batched over t = 0..10)
    wmma_gemm_f32<false, false><<<dim3(32, 1, TM1), 128, 0, stream>>>(
        g, sc, Y1, NODES, BD, NODES, NODES, BD, BD, 1.0f, 0, NBD, NBD);
    wmma_gemm_f32<false, false><<<dim3(32, 1, TM1), 128, 0, stream>>>(
        g, Y1, Y2, NODES, BD, NODES, NODES, BD, BD, 1.0f, 0, NBD, NBD);

    // 6. masked-MAE metrics, argmin, context[0..10]
    metrics_kernel<<<dim3(PER, TM1), 256, 0, stream>>>(sc, Y1, Y2, coeff, metric);
    argmin_kernel<<<1, 32, 0, stream>>>(metric, best);
    build_context_kernel<<<dim3(NN / 256, 1, TM1), 256, 0, stream>>>(
        g, g2, coeff, best, out);

    // 7. cov of sc[T-1] -> context[11]
    xc_kernel<<<NODES, 256, 0, stream>>>(sc + (size_t)(TT - 1) * NBD, xc);
    wmma_gemm_f32<true, false><<<dim3(64), 128, 0, stream>>>(
        xc, xc, out + (size_t)(TT - 1) * NN, NODES, NODES, BD, BD, BD, NODES,
        1.0f / (float)(BD - 1), 0, 0, 0);

    // 8. GRU over flattened context; pick last_idx
    gi_kernel<<<dim3(36, TT), 256, 0, stream>>>(W_ih, out, Gi);
    gru_kernel<<<1, 32, 0, stream>>>(Gi, W_hh, b_ih, b_hh, lastIdx);

    // 9. context[11] = graphs[last_idx]
    build_context_kernel<<<dim3(NN / 256, 1, 1), 256, 0, stream>>>(
        g, g2, coeff, lastIdx, out + (size_t)(TT - 1) * NN);
}